// Sae_43765716746599
// MI455X (gfx1250) — compile-verified
//
#include <hip/hip_runtime.h>
#include <hip/hip_bf16.h>

typedef float v2f __attribute__((ext_vector_type(2)));
typedef float v8f __attribute__((ext_vector_type(8)));

// ---------------- problem dims ----------------
constexpr int Bt   = 2048;    // tokens
constexpr int DIN  = 2048;    // model dim
constexpr int DSAE = 65536;   // latents
constexpr int TOPK = 32;

// ---------------- kernel-1 tiling ----------------
constexpr int MT      = 64;                       // tokens per block
constexpr int NTILE   = 64;                       // latents per chunk
constexpr int NCHUNKS = 32;                       // 2048 latents per block
constexpr int NSPLITS = DSAE / (NTILE * NCHUNKS); // 32
constexpr int MTILES  = Bt / MT;                  // 32
constexpr int KS      = 16;                       // k-slice staged in LDS
constexpr int KSLICES = DIN / KS;                 // 128
constexpr int XS_STRIDE  = 20;                    // pad: conflict-free & 16B-aligned rows
constexpr int LAT_STRIDE = NTILE + 1;             // 65

// ---------------- d_out layout (floats, concat in return order) ----------------
constexpr size_t OUT_SAE  = 0;                           // [Bt][DIN]
constexpr size_t OUT_ACTS = (size_t)Bt * DIN;            // [Bt][TOPK]
constexpr size_t OUT_IDX  = OUT_ACTS + (size_t)Bt*TOPK;  // [Bt][TOPK] (indices as float)
constexpr size_t OUT_FVU  = OUT_IDX + (size_t)Bt*TOPK;   // scalar fvu, then auxk

// ---------------- CDNA5 async copy helpers ----------------
// LDS-relative address = low 32 bits of the generic pointer (ISA 10.2: LDS aperture
// maps flat addr[31:0] -> LDS address).
__device__ __forceinline__ uint32_t lds_u32(const void* p) {
  return (uint32_t)(uintptr_t)p;
}
// GLOBAL_LOAD_ASYNC_TO_LDS_B128: per-lane 16B memory->LDS DMA, tracked by ASYNCcnt.
__device__ __forceinline__ void async_g2l_b128(uint32_t lds_byte, const void* gaddr) {
  asm volatile("global_load_async_to_lds_b128 %0, %1, off"
               :: "v"(lds_byte), "v"(gaddr) : "memory");
}
__device__ __forceinline__ void wait_async0() {
  asm volatile("s_wait_asynccnt 0x0" ::: "memory");
}

// ---------------- kernel 1: fused encoder GEMM (f32 WMMA) + running top-k ----------------
__global__ __launch_bounds__(256) void sae_encode_topk(
    const float* __restrict__ x, const float* __restrict__ W_enc,
    const float* __restrict__ b_enc,
    float* __restrict__ ws_val, int* __restrict__ ws_idx)
{
  __shared__ __attribute__((aligned(16))) float xs[2][MT * XS_STRIDE];    // A staging
  __shared__ __attribute__((aligned(16))) float wl[2][NTILE * XS_STRIDE];// B staging
  __shared__ float lat[MT * LAT_STRIDE];      // post-ReLU chunk
  __shared__ float topv[MT * TOPK];
  __shared__ int   topi[MT * TOPK];

  const int tid  = threadIdx.x;
  const int wave = tid >> 5;
  const int lane = tid & 31;
  const int m0   = blockIdx.x * MT;
  const int nblk = blockIdx.y * (NTILE * NCHUNKS);

  for (int i = tid; i < MT * TOPK; i += 256) { topv[i] = -1.0f; topi[i] = 0; }
  __syncthreads();

  float curMin = -1.0f; int curArg = 0;   // per-token list state (valid for tid < MT)

  // wave -> two 16x16 subtiles of the 64x64 output tile
  const int wm   = wave >> 1;        // row block 0..3 -> rows 16*wm..
  const int wn   = (wave & 1) * 32;  // col base 0 or 32 (two 16-wide subtiles)
  const int l16  = lane & 15;
  const int half = lane >> 4;        // WMMA operand K half (K 0..1 vs 2..3)

  // per-thread staging mapping: one b128 (4 floats of k) per operand per slice
  const int srow = tid >> 2;              // 0..63 (token row / latent row)
  const int skq  = (tid & 3) << 2;        // k offset 0,4,8,12
  const uint32_t xs_lo[2] = { lds_u32(&xs[0][srow * XS_STRIDE + skq]),
                              lds_u32(&xs[1][srow * XS_STRIDE + skq]) };
  const uint32_t wl_lo[2] = { lds_u32(&wl[0][srow * XS_STRIDE + skq]),
                              lds_u32(&wl[1][srow * XS_STRIDE + skq]) };

  for (int c = 0; c < NCHUNKS; ++c) {
    const int nbase = nblk + c * NTILE;
    const float* xg = x + (size_t)m0 * DIN + (size_t)srow * DIN + skq;
    const float* wg = W_enc + (size_t)nbase * DIN + (size_t)srow * DIN + skq;
    v8f acc0 = {}; v8f acc1 = {};

    // stage k-slice 0 via async DMA
    async_g2l_b128(xs_lo[0], xg);
    async_g2l_b128(wl_lo[0], wg);
    wait_async0();
    __syncthreads();

    for (int kb = 0; kb < KSLICES; ++kb) {
      const int cur = kb & 1;
      if (kb + 1 < KSLICES) {               // async prefetch next slice into other buffer
        const int nxt = cur ^ 1;
        const int kof = (kb + 1) * KS;
        async_g2l_b128(xs_lo[nxt], xg + kof);
        async_g2l_b128(wl_lo[nxt], wg + kof);
      }
      if (kb + 2 < KSLICES) {               // L2 prefetch of the HBM-heavy W stream
        __builtin_prefetch(wg + (kb + 2) * KS, 0, 1);
      }
      // 4 WMMA K-steps on current buffer
      const float* xb = xs[cur];
      const float* wb = wl[cur];
      const int arow  = (16 * wm + l16) * XS_STRIDE;
      const int brow0 = (wn + l16) * XS_STRIDE;
      const int brow1 = (wn + 16 + l16) * XS_STRIDE;
      #pragma unroll
      for (int s = 0; s < 4; ++s) {
        const int kc = 4 * s + 2 * half;
        v2f a, b0, b1;
        a.x  = xb[arow + kc];  a.y  = xb[arow + kc + 1];
        b0.x = wb[brow0 + kc]; b0.y = wb[brow0 + kc + 1];
        b1.x = wb[brow1 + kc]; b1.y = wb[brow1 + kc + 1];
        acc0 = __builtin_amdgcn_wmma_f32_16x16x4_f32(false, a, false, b0,
                                                     (short)0, acc0, false, false);
        acc1 = __builtin_amdgcn_wmma_f32_16x16x4_f32(false, a, false, b1,
                                                     (short)0, acc1, false, false);
      }
      wait_async0();      // next slice landed in LDS
      __syncthreads();
    }

    // epilogue: bias + ReLU -> LDS chunk
    {
      const int n0 = wn + l16, n1 = wn + 16 + l16;
      const float bias0 = b_enc[nbase + n0];
      const float bias1 = b_enc[nbase + n1];
      #pragma unroll
      for (int r = 0; r < 8; ++r) {
        const int mrow = 16 * wm + r + half * 8;   // C layout: lanes16-31 hold M+8
        lat[mrow * LAT_STRIDE + n0] = fmaxf(acc0[r] + bias0, 0.0f);
        lat[mrow * LAT_STRIDE + n1] = fmaxf(acc1[r] + bias1, 0.0f);
      }
    }
    __syncthreads();

    // running top-k: one thread per token scans its 64 chunk values
    if (tid < MT) {
      const float* row = lat + tid * LAT_STRIDE;
      float* tv = topv + tid * TOPK;
      int*   ti = topi + tid * TOPK;
      for (int n = 0; n < NTILE; ++n) {
        float v = row[n];
        if (v > curMin) {
          tv[curArg] = v; ti[curArg] = nbase + n;
          float mn = tv[0]; int ag = 0;
          #pragma unroll
          for (int s = 1; s < TOPK; ++s) { float t = tv[s]; if (t < mn) { mn = t; ag = s; } }
          curMin = mn; curArg = ag;
        }
      }
    }
    __syncthreads();
  }

  // write per-(token, n-split) local top-k to workspace
  for (int i = tid; i < MT * TOPK; i += 256) {
    int t = i >> 5, s = i & 31;
    size_t o = (((size_t)(m0 + t) * NSPLITS) + blockIdx.y) * TOPK + s;
    ws_val[o] = topv[i];
    ws_idx[o] = topi[i];
  }
}

// ---------------- kernel 2: merge NSPLITS local top-k -> global sorted top-32 ----------------
__global__ __launch_bounds__(64) void sae_merge_topk(
    const float* __restrict__ ws_val, const int* __restrict__ ws_idx,
    float* __restrict__ out_acts, float* __restrict__ out_idx)
{
  __shared__ float tvs[64 * TOPK];
  __shared__ int   tis[64 * TOPK];
  const int tid = threadIdx.x;
  const int tok = blockIdx.x * 64 + tid;
  float* v  = tvs + tid * TOPK;
  int*   ix = tis + tid * TOPK;
  #pragma unroll
  for (int s = 0; s < TOPK; ++s) { v[s] = -1.0f; ix[s] = 0; }
  float curMin = -1.0f; int curArg = 0;

  const float* cv = ws_val + (size_t)tok * NSPLITS * TOPK;
  const int*   ci = ws_idx + (size_t)tok * NSPLITS * TOPK;
  for (int j = 0; j < NSPLITS * TOPK; ++j) {
    float val = cv[j];
    if (val > curMin) {
      v[curArg] = val; ix[curArg] = ci[j];
      float mn = v[0]; int ag = 0;
      #pragma unroll
      for (int s = 1; s < TOPK; ++s) { float t = v[s]; if (t < mn) { mn = t; ag = s; } }
      curMin = mn; curArg = ag;
    }
  }
  // insertion sort, descending (matches jax.lax.top_k order)
  for (int i = 1; i < TOPK; ++i) {
    float kv = v[i]; int ki = ix[i];
    int j = i - 1;
    while (j >= 0 && v[j] < kv) { v[j+1] = v[j]; ix[j+1] = ix[j]; --j; }
    v[j+1] = kv; ix[j+1] = ki;
  }
  for (int s = 0; s < TOPK; ++s) {
    float val = v[s] < 0.0f ? 0.0f : v[s];      // scrub init sentinels (ReLU outputs are >= 0)
    out_acts[(size_t)tok * TOPK + s] = val;
    out_idx [(size_t)tok * TOPK + s] = (float)ix[s];
  }
}

// ---------------- kernel 3: sparse decode (gather W_dec rows) ----------------
__global__ __launch_bounds__(256) void sae_decode(
    const float* __restrict__ acts, const float* __restrict__ idxf,
    const float* __restrict__ W_dec, const float* __restrict__ b_dec,
    float* __restrict__ out)
{
  __shared__ float a_s[TOPK];
  __shared__ int   i_s[TOPK];
  const int tok = blockIdx.x;
  const int tid = threadIdx.x;
  if (tid < TOPK) {
    a_s[tid] = acts[(size_t)tok * TOPK + tid];
    i_s[tid] = (int)idxf[(size_t)tok * TOPK + tid];  // exact small ints in fp32
  }
  __syncthreads();
  float acc[DIN / 256];
  #pragma unroll
  for (int j = 0; j < DIN / 256; ++j) acc[j] = b_dec[tid + j * 256];
  for (int k = 0; k < TOPK; ++k) {
    const float a = a_s[k];
    const float* wr = W_dec + (size_t)i_s[k] * DIN;
    #pragma unroll
    for (int j = 0; j < DIN / 256; ++j) acc[j] += a * wr[tid + j * 256];
  }
  #pragma unroll
  for (int j = 0; j < DIN / 256; ++j) out[(size_t)tok * DIN + tid + j * 256] = acc[j];
}

// ---------------- kernel 4: per-column FVU ratio ----------------
__global__ __launch_bounds__(256) void sae_fvu_cols(
    const float* __restrict__ x, const float* __restrict__ sae_out,
    float* __restrict__ ratios)
{
  const int j = blockIdx.x * 256 + threadIdx.x;   // column
  float sx = 0.0f, sxx = 0.0f, se = 0.0f;
  for (int b = 0; b < Bt; ++b) {
    float xv = x[(size_t)b * DIN + j];
    float ov = sae_out[(size_t)b * DIN + j];
    sx += xv; sxx += xv * xv;
    float d = ov - xv; se += d * d;
  }
  float tv = sxx - sx * sx / (float)Bt;   // sum((x-mean)^2)
  ratios[j] = se / tv;
}

// ---------------- kernel 5: final reduction -> fvu, auxk ----------------
__global__ __launch_bounds__(256) void sae_fvu_reduce(
    const float* __restrict__ ratios, float* __restrict__ out_fvu)
{
  __shared__ float red[256];
  float s = 0.0f;
  for (int j = threadIdx.x; j < DIN; j += 256) s += ratios[j];
  red[threadIdx.x] = s;
  __syncthreads();
  for (int off = 128; off > 0; off >>= 1) {
    if (threadIdx.x < off) red[threadIdx.x] += red[threadIdx.x + off];
    __syncthreads();
  }
  if (threadIdx.x == 0) { out_fvu[0] = red[0] / (float)DIN; out_fvu[1] = 0.0f; }
}

// ---------------- launch ----------------
extern "C" void kernel_launch(void* const* d_in, const int* in_sizes, int n_in,
                              void* d_out, int out_size, void* d_ws, size_t ws_size,
                              hipStream_t stream) {
  const float* x     = (const float*)d_in[0];
  const float* W_enc = (const float*)d_in[1];
  const float* b_enc = (const float*)d_in[2];
  const float* W_dec = (const float*)d_in[3];
  const float* b_dec = (const float*)d_in[4];
  // d_in[5] is k (==32), baked in as TOPK.

  float* out = (float*)d_out;
  // ws: [Bt*NSPLITS*TOPK] f32 values | [Bt*NSPLITS*TOPK] i32 indices | [DIN] f32 ratios  (~16.8 MB)
  float* ws_val = (float*)d_ws;
  int*   ws_idx = (int*)((char*)d_ws + sizeof(float) * (size_t)Bt * NSPLITS * TOPK);
  float* ws_rat = (float*)((char*)d_ws + 2 * sizeof(float) * (size_t)Bt * NSPLITS * TOPK);

  sae_encode_topk<<<dim3(MTILES, NSPLITS), 256, 0, stream>>>(x, W_enc, b_enc, ws_val, ws_idx);
  sae_merge_topk <<<Bt / 64, 64, 0, stream>>>(ws_val, ws_idx, out + OUT_ACTS, out + OUT_IDX);
  sae_decode     <<<Bt, 256, 0, stream>>>(out + OUT_ACTS, out + OUT_IDX, W_dec, b_dec, out + OUT_SAE);
  sae_fvu_cols   <<<DIN / 256, 256, 0, stream>>>(x, out + OUT_SAE, ws_rat);
  sae_fvu_reduce <<<1, 256, 0, stream>>>(ws_rat, out + OUT_FVU);
}